// MultiHeadAttention_7954279432294
// MI455X (gfx1250) — compile-verified
//
#include <hip/hip_runtime.h>

// MI455X / gfx1250: linear (softmax-free) attention, restructured as
// out = Qp @ (Kp^T @ Vp) to cut attention FLOPs 8x and stay HBM-bound.
// All matmuls via V_WMMA_F32_16X16X4_F32 (full fp32 precision).

typedef __attribute__((ext_vector_type(2))) float v2f;
typedef __attribute__((ext_vector_type(8))) float v8f;

namespace {

constexpr int kB = 8, kC = 64, kH = 64, kW = 512;
constexpr int kHW = kH * kW;      // channel stride in NCHW
constexpr int kWaves = 8;
constexpr int kChunk = 64;        // W columns handled per wave

__device__ __forceinline__ v8f wmma4(v2f a, v2f b, v8f c) {
  // D = A(16x4,f32) * B(4x16,f32) + C(16x16,f32)
  return __builtin_amdgcn_wmma_f32_16x16x4_f32(
      /*neg_a=*/false, a, /*neg_b=*/false, b,
      /*c_mod=*/(short)0, c, /*reuse_a=*/false, /*reuse_b=*/false);
}

// P(64 x 64) = Wm(64x64, LDS) @ X(64 x 64chunk, global NCHW slab) + bias
// dst is a per-wave LDS tile, row-major with ld = kChunk.
__device__ __forceinline__ void project_chunk(
    const float* __restrict__ wm,    // [64x64] LDS, row-major (o,i)
    const float* __restrict__ bias,  // [64] LDS
    const float* __restrict__ xg,    // global input base (NCHW)
    size_t gbase, int col0,
    float* __restrict__ dst, int lane) {
  const int row16 = lane & 15;
  const int cidx  = lane & 15;
  const int hi    = lane >> 4;
  const int kk    = hi * 2;
#pragma unroll
  for (int nw = 0; nw < kChunk / 16; ++nw) {
    // B fragments for the whole K=64 of this 16-column group, straight from
    // global memory in WMMA B layout (half-wave reads 64B contiguous).
    v2f bf[16];
#pragma unroll
    for (int kb = 0; kb < 16; ++kb) {
      const size_t g = gbase + (size_t)(kb * 4 + kk) * kHW +
                       (size_t)(col0 + nw * 16 + cidx);
      bf[kb].x = xg[g];
      bf[kb].y = xg[g + kHW];
    }
#pragma unroll
    for (int mo = 0; mo < 4; ++mo) {
      v8f acc = {0.f, 0.f, 0.f, 0.f, 0.f, 0.f, 0.f, 0.f};
#pragma unroll
      for (int kb = 0; kb < 16; ++kb) {
        v2f a;
        a.x = wm[(mo * 16 + row16) * kC + kb * 4 + kk];
        a.y = wm[(mo * 16 + row16) * kC + kb * 4 + kk + 1];
        acc = wmma4(a, bf[kb], acc);
      }
#pragma unroll
      for (int i = 0; i < 8; ++i) {
        const int r = mo * 16 + i + hi * 8;
        dst[r * kChunk + nw * 16 + cidx] = acc[i] + bias[r];
      }
    }
  }
}

__global__ __launch_bounds__(256, 1)
void mha_linear_attn_kernel(const float* __restrict__ q,
                            const float* __restrict__ k,
                            const float* __restrict__ v,
                            const float* __restrict__ Wq,
                            const float* __restrict__ bq,
                            const float* __restrict__ Wk,
                            const float* __restrict__ bk,
                            const float* __restrict__ Wv,
                            const float* __restrict__ bv,
                            float* __restrict__ out) {
  __shared__ float s_wm[2][kC * kC];             // 32 KB  (Wk,Wv -> Wq reuses [0])
  __shared__ float s_bias[3][kC];                // 768 B  (bq,bk,bv)
  __shared__ float s_S[kC * kC];                 // 16 KB  S = Kp^T Vp
  __shared__ float s_buf0[kWaves][kC * kChunk];  // 128 KB KP, later QP
  __shared__ float s_buf1[kWaves][kC * kChunk];  // 128 KB VP

  const int tid  = threadIdx.x;
  const int wave = tid >> 5;
  const int lane = tid & 31;
  const int row16 = lane & 15;
  const int cidx  = lane & 15;
  const int hi    = lane >> 4;
  const int kk    = hi * 2;

  const int bIdx = blockIdx.x / kH;
  const int hIdx = blockIdx.x % kH;
  const size_t gbase = (size_t)bIdx * kC * kHW + (size_t)hIdx * kW;
  const int col0 = wave * kChunk;

  // Stage weights/biases, zero S.
  for (int i = tid; i < kC * kC; i += 256) {
    s_wm[0][i] = Wk[i];
    s_wm[1][i] = Wv[i];
    s_S[i] = 0.f;
  }
  for (int i = tid; i < kC; i += 256) {
    s_bias[0][i] = bq[i];
    s_bias[1][i] = bk[i];
    s_bias[2][i] = bv[i];
  }
  __syncthreads();

  // ---------------- Phase A: KP, VP, S += KP @ VP^T -----------------------
  float* kp = s_buf0[wave];
  float* vp = s_buf1[wave];
  project_chunk(s_wm[0], s_bias[1], k, gbase, col0, kp, lane);
  project_chunk(s_wm[1], s_bias[2], v, gbase, col0, vp, lane);

  // S_partial[o1,o2] = sum_w KP[o1,w] * VP[o2,w]  (K dim = this wave's 64 w's)
#pragma unroll
  for (int mo = 0; mo < 4; ++mo) {
#pragma unroll
    for (int no = 0; no < 4; ++no) {
      v8f acc = {0.f, 0.f, 0.f, 0.f, 0.f, 0.f, 0.f, 0.f};
#pragma unroll
      for (int kb = 0; kb < 16; ++kb) {
        v2f a;  // A[m,k] = KP[mo*16+m, kb*4+k]
        a.x = kp[(mo * 16 + row16) * kChunk + kb * 4 + kk];
        a.y = kp[(mo * 16 + row16) * kChunk + kb * 4 + kk + 1];
        v2f b;  // B[k,n] = VP^T[k,n] = VP[no*16+n, kb*4+k]
        b.x = vp[(no * 16 + cidx) * kChunk + kb * 4 + kk];
        b.y = vp[(no * 16 + cidx) * kChunk + kb * 4 + kk + 1];
        acc = wmma4(a, b, acc);
      }
#pragma unroll
      for (int i = 0; i < 8; ++i) {
        atomicAdd(&s_S[(mo * 16 + i + hi * 8) * kC + no * 16 + cidx], acc[i]);
      }
    }
  }
  __syncthreads();

  // Swap in Wq (all Phase-A consumers of s_wm[0] are past the barrier).
  for (int i = tid; i < kC * kC; i += 256) s_wm[0][i] = Wq[i];
  __syncthreads();

  // ---------------- Phase B: QP, OUT = S^T @ QP + residual ----------------
  float* qp = s_buf0[wave];  // reuse KP buffer
  project_chunk(s_wm[0], s_bias[0], q, gbase, col0, qp, lane);

  // OUT[c,w] = sum_{c'} S[c',c] * QP[c',w];  then += q (residual), NCHW store
#pragma unroll
  for (int mo = 0; mo < 4; ++mo) {
#pragma unroll
    for (int nw = 0; nw < 4; ++nw) {
      v8f acc = {0.f, 0.f, 0.f, 0.f, 0.f, 0.f, 0.f, 0.f};
#pragma unroll
      for (int kb = 0; kb < 16; ++kb) {
        v2f a;  // A[m,k] = S^T[m,k] = S[kb*4+k, mo*16+m]
        a.x = s_S[(kb * 4 + kk) * kC + mo * 16 + row16];
        a.y = s_S[(kb * 4 + kk + 1) * kC + mo * 16 + row16];
        v2f b;  // B[k,n] = QP[kb*4+k, nw*16+n]
        b.x = qp[(kb * 4 + kk) * kChunk + nw * 16 + cidx];
        b.y = qp[(kb * 4 + kk + 1) * kChunk + nw * 16 + cidx];
        acc = wmma4(a, b, acc);
      }
#pragma unroll
      for (int i = 0; i < 8; ++i) {
        const int c = mo * 16 + i + hi * 8;
        const size_t g = gbase + (size_t)c * kHW +
                         (size_t)(col0 + nw * 16 + cidx);
        out[g] = acc[i] + q[g];  // residual add
      }
    }
  }
}

}  // namespace

extern "C" void kernel_launch(void* const* d_in, const int* in_sizes, int n_in,
                              void* d_out, int out_size, void* d_ws,
                              size_t ws_size, hipStream_t stream) {
  (void)in_sizes; (void)n_in; (void)d_ws; (void)ws_size; (void)out_size;
  const float* q  = (const float*)d_in[0];
  const float* k  = (const float*)d_in[1];
  const float* v  = (const float*)d_in[2];
  const float* Wq = (const float*)d_in[3];
  const float* bq = (const float*)d_in[4];
  const float* Wk = (const float*)d_in[5];
  const float* bk = (const float*)d_in[6];
  const float* Wv = (const float*)d_in[7];
  const float* bv = (const float*)d_in[8];
  float* out = (float*)d_out;

  dim3 grid(kB * kH);   // one workgroup per (b,h) pair: 512
  dim3 block(256);      // 8 wave32 waves, one 64-column W chunk each
  hipLaunchKernelGGL(mha_linear_attn_kernel, grid, block, 0, stream,
                     q, k, v, Wq, bq, Wk, bk, Wv, bv, out);
}